// rootfind_module_77309411649
// MI455X (gfx1250) — compile-verified
//
#include <hip/hip_runtime.h>
#include <hip/hip_bf16.h>

typedef __attribute__((ext_vector_type(2))) float v2f;
typedef __attribute__((ext_vector_type(4))) float v4f;
typedef __attribute__((ext_vector_type(8))) float v8f;

// Problem constants (match reference setup_inputs)
#define BB   8192
#define DD   512
#define HH   2048
#define HV   512
#define VO   256
#define NEWTON_ITERS 50

// ---------------------------------------------------------------------------
// Tiled f32 GEMM via V_WMMA_F32_16X16X4_F32.
// C[M,N] = op(A[M,K] @ B[K,N] + bias), op = tanh if mode==1.
// Block = 128 threads = 4 wave32 (2x2); each wave computes a 32x32 register
// tile (4 x v8f accumulators) of a 64x64 workgroup tile.
// K staged in chunks of 32 through DOUBLE-BUFFERED LDS:
//   As[buf][64][36]  row-major  (stride 36: b64-aligned, conflict-free frags)
//   Bs[buf][64][34]  K-transposed [n][k] (stride 34: b64-aligned frags,
//                    contiguous {K,K+1} pairs -> no VGPR assembly movs)
// Loop: store staged registers to LDS buffer buf; ONE barrier; issue next chunk's b128 global
// loads (overlap compute); 8 k-steps x 4 back-to-back WMMAs.
// Fragment layouts per CDNA5 ISA 7.12.2:
//   A (16x4 f32): lane m=l&15, half=l>>4 -> {K=2*half, K=2*half+1}
//   B (4x16 f32): same K striping, N = l&15
//   C/D (16x16 f32, 8 VGPRs): vgpr r -> row r + 8*(l>>4), col = l&15
// M,N,K multiples of 64/64/32 for every call in this pipeline.
// ---------------------------------------------------------------------------
__global__ __launch_bounds__(128)
void wmma_gemm_f32(const float* __restrict__ A, const float* __restrict__ B,
                   const float* __restrict__ bias, float* __restrict__ C,
                   int M, int N, int K, int mode) {
    __shared__ float As[2][64][36];
    __shared__ float Bs[2][64][34];   // [n][k] transposed

    const int gn0  = blockIdx.x * 64;
    const int gm0  = blockIdx.y * 64;
    const int tid  = threadIdx.x;
    const int lane = tid & 31;
    const int wave = tid >> 5;
    const int wm   = (wave >> 1) * 32;   // wave row offset in WG tile
    const int wn   = (wave & 1) * 32;    // wave col offset in WG tile
    const int ml   = lane & 15;
    const int kh   = lane >> 4;

    v8f acc00 = {0.f,0.f,0.f,0.f,0.f,0.f,0.f,0.f};
    v8f acc01 = acc00, acc10 = acc00, acc11 = acc00;

    const int ia = tid >> 3, ja = (tid & 7) * 4;    // A: 64 rows x 32 cols
    const int ib = tid >> 4, jb = (tid & 15) * 4;   // B: 32 rows x 64 cols

    v4f ra[4], rb[4];
    // prologue: load chunk 0
#pragma unroll
    for (int p = 0; p < 4; ++p) {
        ra[p] = *(const v4f*)&A[(size_t)(gm0 + ia + p * 16) * K + ja];
        rb[p] = *(const v4f*)&B[(size_t)(ib + p * 8) * N + (gn0 + jb)];
    }

    const int nchunks = K >> 5;
    for (int c = 0; c < nchunks; ++c) {
        const int buf = c & 1;
        // ---- store staged chunk to LDS (B transposed) ----
#pragma unroll
        for (int p = 0; p < 4; ++p) {
            *(v4f*)&As[buf][ia + p * 16][ja] = ra[p];
            Bs[buf][jb + 0][ib + p * 8] = rb[p][0];
            Bs[buf][jb + 1][ib + p * 8] = rb[p][1];
            Bs[buf][jb + 2][ib + p * 8] = rb[p][2];
            Bs[buf][jb + 3][ib + p * 8] = rb[p][3];
        }
        __syncthreads();

        // ---- issue next chunk's global loads (overlap with compute) ----
        if (c + 1 < nchunks) {
            const int kb = (c + 1) * 32;
#pragma unroll
            for (int p = 0; p < 4; ++p) {
                ra[p] = *(const v4f*)&A[(size_t)(gm0 + ia + p * 16) * K + (kb + ja)];
                rb[p] = *(const v4f*)&B[(size_t)(kb + ib + p * 8) * N + (gn0 + jb)];
            }
        }

        // ---- compute: 8 k-steps x 4 WMMA ----
#pragma unroll
        for (int k = 0; k < 32; k += 4) {
            v2f a0 = *(const v2f*)&As[buf][wm + ml][k + 2 * kh];
            v2f a1 = *(const v2f*)&As[buf][wm + 16 + ml][k + 2 * kh];
            v2f b0 = *(const v2f*)&Bs[buf][wn + ml][k + 2 * kh];
            v2f b1 = *(const v2f*)&Bs[buf][wn + 16 + ml][k + 2 * kh];
            acc00 = __builtin_amdgcn_wmma_f32_16x16x4_f32(false, a0, false, b0, (short)0, acc00, false, false);
            acc01 = __builtin_amdgcn_wmma_f32_16x16x4_f32(false, a0, false, b1, (short)0, acc01, false, false);
            acc10 = __builtin_amdgcn_wmma_f32_16x16x4_f32(false, a1, false, b0, (short)0, acc10, false, false);
            acc11 = __builtin_amdgcn_wmma_f32_16x16x4_f32(false, a1, false, b1, (short)0, acc11, false, false);
        }
    }

    // ---- epilogue: optional bias add + tanh ----
#pragma unroll
    for (int mi = 0; mi < 2; ++mi) {
#pragma unroll
        for (int ni = 0; ni < 2; ++ni) {
            v8f acc = (mi == 0) ? (ni == 0 ? acc00 : acc01)
                                : (ni == 0 ? acc10 : acc11);
            int colb = gn0 + wn + ni * 16 + ml;
#pragma unroll
            for (int r = 0; r < 8; ++r) {
                int row = gm0 + wm + mi * 16 + r + 8 * kh;
                float v = acc[r];
                if (bias) v += bias[colb];
                if (mode) v = tanhf(v);
                C[(size_t)row * N + colb] = v;
            }
        }
    }
}

// ---------------- wave32 reductions ----------------
__device__ inline float wave_reduce_add(float v) {
#pragma unroll
    for (int o = 16; o > 0; o >>= 1) v += __shfl_down(v, o, 32);
    return v;
}

// 256-thread block reduction; result valid in thread 0
__device__ inline float block_reduce_add(float v, float* sm) {
    const int lane = threadIdx.x & 31, w = threadIdx.x >> 5;
    v = wave_reduce_add(v);
    if (lane == 0) sm[w] = v;
    __syncthreads();
    float s = 0.f;
    if (w == 0) {
        s = (lane < 8) ? sm[lane] : 0.f;
        s = wave_reduce_add(s);
    }
    return s;
}

// target[m] = 0.99 * ( sum_j yx[m,j]^2 + 0.01 * sum_d x[m,d]^2 )
__global__ __launch_bounds__(256)
void target_kernel(const float* __restrict__ yx, const float* __restrict__ x,
                   float* __restrict__ target) {
    __shared__ float sm[8];
    const int m = blockIdx.x, t = threadIdx.x;
    float yv = yx[(size_t)m * VO + t];
    float x0 = x[(size_t)m * DD + t];
    float x1 = x[(size_t)m * DD + 256 + t];
    float s = block_reduce_add(yv * yv + 0.01f * (x0 * x0 + x1 * x1), sm);
    if (t == 0) target[m] = 0.99f * s;
}

// fnorm[m] = sum_d fhatx[m,d]^2 ; alpha[m] = 1
__global__ __launch_bounds__(256)
void fnorm_init_kernel(const float* __restrict__ fhatx,
                       float* __restrict__ fnorm, float* __restrict__ alpha) {
    __shared__ float sm[8];
    const int m = blockIdx.x, t = threadIdx.x;
    float f0 = fhatx[(size_t)m * DD + t];
    float f1 = fhatx[(size_t)m * DD + 256 + t];
    float s = block_reduce_add(f0 * f0 + f1 * f1, sm);
    if (t == 0) { fnorm[m] = s; alpha[m] = 1.0f; }
}

// Hm = tanh(alpha[m]*g), Hp = (1-Hm^2)*g   (elementwise over B*HV)
__global__ __launch_bounds__(256)
void newton_prep_kernel(const float* __restrict__ g,
                        const float* __restrict__ alpha,
                        float* __restrict__ Hm, float* __restrict__ Hp) {
    size_t idx = (size_t)blockIdx.x * 256 + threadIdx.x;
    int m = (int)(idx >> 9);  // HV == 512
    float gg = g[idx];
    float h = tanhf(alpha[m] * gg);
    Hm[idx] = h;
    Hp[idx] = (1.0f - h * h) * gg;
}

// V = |y|^2 + 0.01*a^2*fn ; dVda = 2*(y.dy) + 0.02*a*fn ;
// alpha -= resid/dVda only where resid > 1e-3 (masked Newton, matches ref)
__global__ __launch_bounds__(256)
void newton_update_kernel(const float* __restrict__ y,
                          const float* __restrict__ dy,
                          const float* __restrict__ target,
                          const float* __restrict__ fnorm,
                          float* __restrict__ alpha) {
    __shared__ float smv[8];
    __shared__ float smd[8];
    const int m = blockIdx.x, t = threadIdx.x;
    float yv = y[(size_t)m * VO + t];
    float dv = dy[(size_t)m * VO + t];
    float sv = block_reduce_add(yv * yv, smv);
    __syncthreads();
    float sd = block_reduce_add(yv * dv, smd);
    if (t == 0) {
        float a  = alpha[m];
        float fn = fnorm[m];
        float V    = sv + 0.01f * a * a * fn;
        float dVda = 2.0f * sd + 0.02f * a * fn;
        float resid = V - target[m];
        if (resid > 1e-3f) alpha[m] = a - resid / dVda;
    }
}

// out = fhatx * alpha (broadcast per sample), float4 vectorized
__global__ __launch_bounds__(256)
void finalize_kernel(const float* __restrict__ fhatx,
                     const float* __restrict__ alpha,
                     float* __restrict__ out) {
    size_t q = (size_t)blockIdx.x * 256 + threadIdx.x;   // float4 index
    int m = (int)(q >> 7);                               // DD/4 == 128
    v4f f = *(const v4f*)&fhatx[q * 4];
    float a = alpha[m];
    v4f o = {f[0] * a, f[1] * a, f[2] * a, f[3] * a};
    *(v4f*)&out[q * 4] = o;
}

extern "C" void kernel_launch(void* const* d_in, const int* in_sizes, int n_in,
                              void* d_out, int out_size, void* d_ws, size_t ws_size,
                              hipStream_t stream) {
    (void)in_sizes; (void)n_in; (void)out_size; (void)ws_size;

    const float* x   = (const float*)d_in[0];
    const float* W1  = (const float*)d_in[1];
    const float* b1  = (const float*)d_in[2];
    const float* W2  = (const float*)d_in[3];
    const float* b2  = (const float*)d_in[4];
    const float* Wv1 = (const float*)d_in[5];
    const float* Wv2 = (const float*)d_in[6];
    float* out = (float*)d_out;

    // Workspace layout (floats)
    float* ws = (float*)d_ws;
    const size_t BUF0   = 0;                                // B*H scratch, reused
    const size_t FHATX  = BUF0 + (size_t)BB * HH;           // B*D
    const size_t G      = FHATX + (size_t)BB * DD;          // B*HV
    const size_t TARGET = G + (size_t)BB * HV;              // B
    const size_t FNORM  = TARGET + BB;                      // B
    const size_t ALPHA  = FNORM + BB;                       // B

    float* h1    = ws + BUF0;                  // B x H
    float* fhatx = ws + FHATX;                 // B x D
    float* g     = ws + G;                     // B x HV
    float* tgt   = ws + TARGET;
    float* fnorm = ws + FNORM;
    float* alpha = ws + ALPHA;
    // sub-buffers carved out of BUF0 after h1 is consumed
    float* Hm = ws + BUF0;                       // B x HV
    float* Hp = ws + BUF0 + (size_t)BB * HV;     // B x HV
    float* y  = ws + BUF0 + (size_t)2 * BB * HV; // B x VO
    float* dy = ws + BUF0 + (size_t)2 * BB * HV + (size_t)BB * VO; // B x VO
    float* hx = Hm;                              // B x HV (reuse)
    float* yx = y;                               // B x VO (reuse)

    dim3 blk(128);

    // 1) h1 = tanh(x @ W1 + b1)        [8192 x 2048, K=512]
    wmma_gemm_f32<<<dim3(HH / 64, BB / 64), blk, 0, stream>>>(x, W1, b1, h1, BB, HH, DD, 1);
    // 2) fhatx = h1 @ W2 + b2          [8192 x 512, K=2048]
    wmma_gemm_f32<<<dim3(DD / 64, BB / 64), blk, 0, stream>>>(h1, W2, b2, fhatx, BB, DD, HH, 0);
    // 3) hx = tanh(x @ Wv1)            [8192 x 512, K=512]  (h1 now dead)
    wmma_gemm_f32<<<dim3(HV / 64, BB / 64), blk, 0, stream>>>(x, Wv1, nullptr, hx, BB, HV, DD, 1);
    // 4) yx = hx @ Wv2                 [8192 x 256, K=512]
    wmma_gemm_f32<<<dim3(VO / 64, BB / 64), blk, 0, stream>>>(hx, Wv2, nullptr, yx, BB, VO, HV, 0);
    // 5) target = 0.99 * V(x)
    target_kernel<<<BB, 256, 0, stream>>>(yx, x, tgt);
    // 6) g = fhatx @ Wv1 (alpha-independent)
    wmma_gemm_f32<<<dim3(HV / 64, BB / 64), blk, 0, stream>>>(fhatx, Wv1, nullptr, g, BB, HV, DD, 0);
    // 7) fnorm = |fhatx|^2, alpha = 1
    fnorm_init_kernel<<<BB, 256, 0, stream>>>(fhatx, fnorm, alpha);

    // 8) Newton loop (fixed trip count with per-sample freeze == masked while)
    const int ewBlocks = (BB * HV) / 256;
    for (int it = 0; it < NEWTON_ITERS; ++it) {
        newton_prep_kernel<<<ewBlocks, 256, 0, stream>>>(g, alpha, Hm, Hp);
        wmma_gemm_f32<<<dim3(VO / 64, BB / 64), blk, 0, stream>>>(Hm, Wv2, nullptr, y, BB, VO, HV, 0);
        wmma_gemm_f32<<<dim3(VO / 64, BB / 64), blk, 0, stream>>>(Hp, Wv2, nullptr, dy, BB, VO, HV, 0);
        newton_update_kernel<<<BB, 256, 0, stream>>>(y, dy, tgt, fnorm, alpha);
    }

    // 9) out = fhatx * alpha
    finalize_kernel<<<(BB * DD) / (256 * 4), 256, 0, stream>>>(fhatx, alpha, out);
}